// GNN_0_24567212933213
// MI455X (gfx1250) — compile-verified
//
#include <hip/hip_runtime.h>
#include <hip/hip_bf16.h>

// ---------------------------------------------------------------------------
// GCN forward on gfx1250 (CDNA5), fp32 end-to-end.
//   deg/dinv -> [GEMM(WMMA f32, LDS-staged) -> self-loop init -> edge scatter
//   -> bias/relu] x3 -> mean pool -> 64x2 linear
// Matrix math: V_WMMA_F32_16X16X4_F32, one 16x16 C tile per wave32,
// fully unrolled K loop, all fragments fed by ds_load_b64.
// ---------------------------------------------------------------------------

typedef __attribute__((ext_vector_type(2))) float v2f;
typedef __attribute__((ext_vector_type(8))) float v8f;

#define N_NODES  100000
#define N_GRAPHS 1024
#define IN_DIM   50
#define HID      128
#define OUT3     64

// ---------------- degree / normalization ----------------

__global__ void gcn_init_deg(float* __restrict__ deg, int n) {
    int i = blockIdx.x * blockDim.x + threadIdx.x;
    if (i < n) deg[i] = 1.0f;  // self-loop contributes 1
}

__global__ void gcn_count_deg(const int* __restrict__ dst, float* __restrict__ deg, int e) {
    int i = blockIdx.x * blockDim.x + threadIdx.x;
    if (i < e) atomicAdd(&deg[dst[i]], 1.0f);
}

__global__ void gcn_make_dinv(float* __restrict__ deg, int n) {
    int i = blockIdx.x * blockDim.x + threadIdx.x;
    if (i < n) deg[i] = rsqrtf(fmaxf(deg[i], 1.0f));  // in place: deg -> dinv
}

// ---------------- dense GEMM via V_WMMA_F32_16X16X4_F32 ----------------
// H[M,N] = X[M,K] @ W[K,N].  Compile-time K,N.  One 16x16 C tile per wave;
// wave w covers cols [16w,16w+16).  blockDim.x = 32*(N/16).  M % 16 == 0
// (100000 = 6250*16).  A tile and W staged in LDS, zero-padded to KP, so the
// unrolled inner loop is pure ds_load_b64 + wmma with uniform EXEC.

template <int K, int N>
__global__ __launch_bounds__(256) void gcn_gemm_wmma(
    const float* __restrict__ X, const float* __restrict__ W,
    float* __restrict__ H) {
    constexpr int KP = (K + 3) & ~3;                 // K padded to multiple of 4
    __shared__ __align__(16) float sW[N * KP];       // col-major: sW[n*KP + k]
    __shared__ __align__(16) float sA[16 * KP];      // row-major: sA[r*KP + k]

    const int tid = threadIdx.x;
    const int m0  = blockIdx.x * 16;

    // stage weights (coalesced global read, transpose into LDS, zero-pad tail)
    for (int idx = tid; idx < KP * N; idx += blockDim.x) {
        const int k = idx / N, n = idx % N;
        sW[n * KP + k] = (k < K) ? W[idx] : 0.0f;
    }
    // stage 16-row A tile, zero-pad tail
    for (int idx = tid; idx < 16 * KP; idx += blockDim.x) {
        const int r = idx / KP, k = idx % KP;
        sA[idx] = (k < K) ? X[(long long)(m0 + r) * K + k] : 0.0f;
    }
    __syncthreads();

    const int lane = tid & 31;
    const int wid  = tid >> 5;
    const int n0   = wid * 16;
    const int half = lane >> 4;   // 0: lanes 0-15, 1: lanes 16-31
    const int l16  = lane & 15;
    const int col  = n0 + l16;

    // ISA A layout (16x4 f32): lanes 0-15 hold K={kk,kk+1}, lanes 16-31 {kk+2,kk+3}
    const v2f* pa = (const v2f*)&sA[l16 * KP];   // 8B-aligned: KP%4==0, kb even
    const v2f* pb = (const v2f*)&sW[col * KP];

    v8f acc = {};
#pragma unroll
    for (int kk = 0; kk < KP; kk += 4) {
        const int kb2 = (kk >> 1) + half;        // v2f index of element kk+2*half
        v2f a = pa[kb2];
        v2f b = pb[kb2];
        // (neg_a, A, neg_b, B, c_mod, C, reuse_a, reuse_b)
        acc = __builtin_amdgcn_wmma_f32_16x16x4_f32(
            false, a, false, b, (short)0, acc, false, false);
    }

    // C layout: VGPR i -> rows m0+i (lanes 0-15) and m0+8+i (lanes 16-31)
#pragma unroll
    for (int i = 0; i < 8; ++i) {
        H[(long long)(m0 + i + 8 * half) * N + col] = acc[i];
    }
}

// ---------------- aggregation: ACC = Ahat @ H (float4 paths) ----------------

// self-loop term doubles as initializer: ACC[i,f] = dinv[i]^2 * H[i,f]
__global__ void gcn_selfloop_init(const float4* __restrict__ H,
                                  const float* __restrict__ dinv,
                                  float4* __restrict__ ACC, int n, int Fq) {
    int idx = blockIdx.x * blockDim.x + threadIdx.x;
    if (idx >= n * Fq) return;
    const int i = idx / Fq;
    const float d = dinv[i];
    const float s = d * d;
    float4 h = H[idx];
    ACC[idx] = make_float4(s * h.x, s * h.y, s * h.z, s * h.w);
}

// one thread per (edge, 4-feature chunk): ACC[dst,f..f+3] += norm * H[src,f..f+3]
__global__ void gcn_edge_scatter(const float4* __restrict__ H,
                                 const float* __restrict__ dinv,
                                 const int* __restrict__ src,
                                 const int* __restrict__ dst,
                                 float* __restrict__ ACC, int e, int Fq) {
    long long t = (long long)blockIdx.x * blockDim.x + threadIdx.x;
    if (t >= (long long)e * Fq) return;
    const int ei = (int)(t / Fq);
    const int q  = (int)(t % Fq);
    const int s = src[ei];
    const int d = dst[ei];
    const float c = dinv[s] * dinv[d];
    float4 h = H[(long long)s * Fq + q];
    float* base = ACC + ((long long)d * Fq + q) * 4;
    atomicAdd(base + 0, c * h.x);
    atomicAdd(base + 1, c * h.y);
    atomicAdd(base + 2, c * h.z);
    atomicAdd(base + 3, c * h.w);
}

__global__ void gcn_bias_act(float4* __restrict__ ACC, const float4* __restrict__ bias,
                             int n, int Fq, int do_relu) {
    int idx = blockIdx.x * blockDim.x + threadIdx.x;
    if (idx >= n * Fq) return;
    const int q = idx % Fq;
    float4 v = ACC[idx];
    float4 b = bias[q];
    v.x += b.x; v.y += b.y; v.z += b.z; v.w += b.w;
    if (do_relu) {
        v.x = fmaxf(v.x, 0.0f); v.y = fmaxf(v.y, 0.0f);
        v.z = fmaxf(v.z, 0.0f); v.w = fmaxf(v.w, 0.0f);
    }
    ACC[idx] = v;
}

// ---------------- pooling + final linear ----------------

__global__ void gcn_pool_zero(float* __restrict__ psum, float* __restrict__ pcnt) {
    int idx = blockIdx.x * blockDim.x + threadIdx.x;
    if (idx < N_GRAPHS * OUT3) psum[idx] = 0.0f;
    if (idx < N_GRAPHS) pcnt[idx] = 0.0f;
}

__global__ void gcn_pool_accum(const float4* __restrict__ H3,
                               const int* __restrict__ batch,
                               float* __restrict__ psum, float* __restrict__ pcnt, int n) {
    constexpr int Fq = OUT3 / 4;
    int idx = blockIdx.x * blockDim.x + threadIdx.x;
    if (idx >= n * Fq) return;
    const int i = idx / Fq;
    const int q = idx % Fq;
    const int g = batch[i];
    float4 h = H3[idx];
    float* base = psum + g * OUT3 + q * 4;
    atomicAdd(base + 0, h.x);
    atomicAdd(base + 1, h.y);
    atomicAdd(base + 2, h.z);
    atomicAdd(base + 3, h.w);
    if (q == 0) atomicAdd(&pcnt[g], 1.0f);
}

__global__ void gcn_final_linear(const float* __restrict__ psum,
                                 const float* __restrict__ pcnt,
                                 const float* __restrict__ Wlin,
                                 const float* __restrict__ blin,
                                 float* __restrict__ out) {
    int t = blockIdx.x * blockDim.x + threadIdx.x;
    if (t >= N_GRAPHS * 2) return;
    const int g = t / 2;
    const int j = t % 2;
    const float inv = 1.0f / fmaxf(pcnt[g], 1.0f);
    float a = blin[j];
#pragma unroll
    for (int f = 0; f < OUT3; ++f)
        a += psum[g * OUT3 + f] * inv * Wlin[f * 2 + j];
    out[t] = a;
}

// ---------------------------------------------------------------------------

extern "C" void kernel_launch(void* const* d_in, const int* in_sizes, int n_in,
                              void* d_out, int out_size, void* d_ws, size_t ws_size,
                              hipStream_t stream) {
    const float* x    = (const float*)d_in[0];
    const int*   eidx = (const int*)d_in[1];
    const int*   batch= (const int*)d_in[2];
    const float* W1   = (const float*)d_in[3];
    const float* b1   = (const float*)d_in[4];
    const float* W2   = (const float*)d_in[5];
    const float* b2   = (const float*)d_in[6];
    const float* W3   = (const float*)d_in[7];
    const float* b3   = (const float*)d_in[8];
    const float* Wlin = (const float*)d_in[9];
    const float* blin = (const float*)d_in[10];
    float* out = (float*)d_out;

    const int n = N_NODES;
    const int e = in_sizes[1] / 2;  // edge_index is (2, E)
    const int* src = eidx;
    const int* dst = eidx + e;

    // workspace layout (floats)
    float* wsf  = (float*)d_ws;
    float* dinv = wsf;                               // n (padded slot: 100352)
    float* bufA = wsf + 100352;                      // n * 128 (GEMM output h)
    float* bufB = bufA + (size_t)n * HID;            // n * 128 (aggregated layer out)
    float* psum = bufB + (size_t)n * HID;            // 1024 * 64
    float* pcnt = psum + N_GRAPHS * OUT3;            // 1024
    (void)ws_size; (void)n_in; (void)out_size;

    const int T = 256;
    auto blks = [](long long work, int t) { return (int)((work + t - 1) / t); };

    // --- degree -> dinv (in place) ---
    gcn_init_deg <<<blks(n, T), T, 0, stream>>>(dinv, n);
    gcn_count_deg<<<blks(e, T), T, 0, stream>>>(dst, dinv, e);
    gcn_make_dinv<<<blks(n, T), T, 0, stream>>>(dinv, n);

    // --- layer 1: h1 = relu(Ahat @ (x @ W1) + b1) ---
    gcn_gemm_wmma<IN_DIM, HID><<<n / 16, 32 * (HID / 16), 0, stream>>>(x, W1, bufA);
    gcn_selfloop_init<<<blks((long long)n * (HID/4), T), T, 0, stream>>>(
        (const float4*)bufA, dinv, (float4*)bufB, n, HID/4);
    gcn_edge_scatter <<<blks((long long)e * (HID/4), T), T, 0, stream>>>(
        (const float4*)bufA, dinv, src, dst, bufB, e, HID/4);
    gcn_bias_act     <<<blks((long long)n * (HID/4), T), T, 0, stream>>>(
        (float4*)bufB, (const float4*)b1, n, HID/4, 1);

    // --- layer 2: h2 = relu(Ahat @ (h1 @ W2) + b2) ---
    gcn_gemm_wmma<HID, HID><<<n / 16, 32 * (HID / 16), 0, stream>>>(bufB, W2, bufA);
    gcn_selfloop_init<<<blks((long long)n * (HID/4), T), T, 0, stream>>>(
        (const float4*)bufA, dinv, (float4*)bufB, n, HID/4);
    gcn_edge_scatter <<<blks((long long)e * (HID/4), T), T, 0, stream>>>(
        (const float4*)bufA, dinv, src, dst, bufB, e, HID/4);
    gcn_bias_act     <<<blks((long long)n * (HID/4), T), T, 0, stream>>>(
        (float4*)bufB, (const float4*)b2, n, HID/4, 1);

    // --- layer 3: h3 = Ahat @ (h2 @ W3) + b3 (no relu) ---
    gcn_gemm_wmma<HID, OUT3><<<n / 16, 32 * (OUT3 / 16), 0, stream>>>(bufB, W3, bufA);
    gcn_selfloop_init<<<blks((long long)n * (OUT3/4), T), T, 0, stream>>>(
        (const float4*)bufA, dinv, (float4*)bufB, n, OUT3/4);
    gcn_edge_scatter <<<blks((long long)e * (OUT3/4), T), T, 0, stream>>>(
        (const float4*)bufA, dinv, src, dst, bufB, e, OUT3/4);
    gcn_bias_act     <<<blks((long long)n * (OUT3/4), T), T, 0, stream>>>(
        (float4*)bufB, (const float4*)b3, n, OUT3/4, 0);

    // --- global mean pool + linear ---
    gcn_pool_zero  <<<blks(N_GRAPHS * OUT3, T), T, 0, stream>>>(psum, pcnt);
    gcn_pool_accum <<<blks((long long)n * (OUT3/4), T), T, 0, stream>>>(
        (const float4*)bufB, batch, psum, pcnt, n);
    gcn_final_linear<<<blks(N_GRAPHS * 2, T), T, 0, stream>>>(psum, pcnt, Wlin, blin, out);
}